// GNN_60490319397059
// MI455X (gfx1250) — compile-verified
//
#include <hip/hip_runtime.h>
#include <hip/hip_bf16.h>

typedef float v2f __attribute__((ext_vector_type(2)));
typedef float v8f __attribute__((ext_vector_type(8)));

#define BDIM 8192
#define NNODE 10000
#define NEDGE 320000
#define FDIM 4396
#define GEPS 1e-5f

// ---------------------------------------------------------------------------
// Tiled WMMA f32 GEMM: C[M,N] = A[M,K] @ B[K,N] (+ bias per column, optional)
// Block tile 64x128, Kc = 32, 256 threads = 8 waves; wave grid 4x2, each wave
// owns a 16x64 strip (4 accumulators) -> 4 WMMAs per A-fragment load.
// LDS layouts give single aligned ds_load_b64 per WMMA operand:
//   A: row stride 34 (pad 2) -> {A[r][k],A[r][k+1]} contiguous, banks distinct
//   B: pair-major Bp[k/2][n] = {B[k][n],B[k+1][n]} -> fragment = one b64 load
// Uses V_WMMA_F32_16X16X4_F32 (wave32, fragments per ISA 7.12.2).
// ---------------------------------------------------------------------------
__global__ __launch_bounds__(256)
void wmma_gemm_f32(const float* __restrict__ A, const float* __restrict__ B,
                   const float* __restrict__ bias, float* __restrict__ C,
                   int M, int N, int K)
{
    __shared__ __align__(16) float Asm[64 * 34];    // 8.5 KB
    __shared__ __align__(16) float Bsm[16 * 256];   // 16 KB (16 k-pairs x 128 cols x 2)

    const int tid  = threadIdx.x;
    const int lane = tid & 31;
    const int wave = tid >> 5;     // 0..7
    const int wm   = wave & 3;     // wave row -> rows wm*16
    const int wn   = wave >> 2;    // wave col -> cols wn*64
    const int bm   = blockIdx.y * 64;
    const int bn   = blockIdx.x * 128;

    const int l16  = lane & 15;
    const int half = lane >> 4;    // 0: lanes 0-15, 1: lanes 16-31

    v8f acc0 = {}, acc1 = {}, acc2 = {}, acc3 = {};

    const bool fullM = (bm + 64 <= M);
    const bool fullN = (bn + 128 <= N);

    for (int k0 = 0; k0 < K; k0 += 32) {
        const bool fullK = (k0 + 32 <= K);

        // ---- stage A tile 64x32 ----
        if (fullM && fullK) {
            #pragma unroll
            for (int i = 0; i < 2; ++i) {               // 512 float4 / 256 thr
                int idx = tid + i * 256;
                int r = idx >> 3, c4 = (idx & 7) << 2;
                const float4 v = *(const float4*)(A + (size_t)(bm + r) * K + k0 + c4);
                float* dst = &Asm[r * 34 + c4];
                dst[0] = v.x; dst[1] = v.y; dst[2] = v.z; dst[3] = v.w;
            }
        } else {
            #pragma unroll
            for (int i = 0; i < 8; ++i) {
                int e = tid + i * 256;
                int r = e >> 5, c = e & 31;
                int gr = bm + r, gc = k0 + c;
                float v = 0.f;
                if (gr < M && gc < K) v = A[(size_t)gr * K + gc];
                Asm[r * 34 + c] = v;
            }
        }

        // ---- stage B tile 32x128 (pair-major) ----
        if (fullN && fullK) {
            #pragma unroll
            for (int i = 0; i < 4; ++i) {               // 1024 float4 / 256 thr
                int idx = tid + i * 256;
                int r = idx >> 5, c4 = (idx & 31) << 2;
                const float4 v = *(const float4*)(B + (size_t)(k0 + r) * N + bn + c4);
                float* dst = &Bsm[(r >> 1) * 256 + (r & 1)];
                dst[(c4 + 0) * 2] = v.x;
                dst[(c4 + 1) * 2] = v.y;
                dst[(c4 + 2) * 2] = v.z;
                dst[(c4 + 3) * 2] = v.w;
            }
        } else {
            #pragma unroll
            for (int i = 0; i < 16; ++i) {
                int e = tid + i * 256;
                int r = e >> 7, c = e & 127;
                int gr = k0 + r, gc = bn + c;
                float v = 0.f;
                if (gr < K && gc < N) v = B[(size_t)gr * N + gc];
                Bsm[(r >> 1) * 256 + c * 2 + (r & 1)] = v;
            }
        }
        __syncthreads();

        const float* ap = &Asm[(wm * 16 + l16) * 34];
        const float* bp = &Bsm[wn * 128 + l16 * 2];
        #pragma unroll
        for (int kk = 0; kk < 32; kk += 4) {
            // A frag 16x4: VGPR pair holds K=(kk,kk+1) | (kk+2,kk+3) by half
            v2f a = *(const v2f*)(ap + kk + half * 2);
            const int prow = ((kk >> 1) + half) * 256;
            v2f b0 = *(const v2f*)(bp + prow + 0);
            v2f b1 = *(const v2f*)(bp + prow + 32);
            v2f b2 = *(const v2f*)(bp + prow + 64);
            v2f b3 = *(const v2f*)(bp + prow + 96);
            acc0 = __builtin_amdgcn_wmma_f32_16x16x4_f32(false, a, false, b0,
                                                         (short)0, acc0, false, false);
            acc1 = __builtin_amdgcn_wmma_f32_16x16x4_f32(false, a, false, b1,
                                                         (short)0, acc1, false, false);
            acc2 = __builtin_amdgcn_wmma_f32_16x16x4_f32(false, a, false, b2,
                                                         (short)0, acc2, false, false);
            acc3 = __builtin_amdgcn_wmma_f32_16x16x4_f32(false, a, false, b3,
                                                         (short)0, acc3, false, false);
        }
        __syncthreads();
    }

    // ---- epilogue: D layout VGPR v -> row (v + 8*half), col l16 ----
    v8f accs[4] = {acc0, acc1, acc2, acc3};
    #pragma unroll
    for (int t = 0; t < 4; ++t) {
        int col = bn + wn * 64 + t * 16 + l16;
        if (col >= N) continue;
        float bv = bias ? bias[col] : 0.f;
        #pragma unroll
        for (int v = 0; v < 8; ++v) {
            int row = bm + wm * 16 + v + half * 8;
            if (row < M) C[(size_t)row * N + col] = accs[t][v] + bv;
        }
    }
}

// ---------------------------------------------------------------------------
// Elementwise / GCN helper kernels
// ---------------------------------------------------------------------------
__global__ void fill_kernel(float* __restrict__ p, float v, int n) {
    int i = blockIdx.x * 256 + threadIdx.x;
    if (i < n) p[i] = v;
}

__global__ void deg_kernel(const int* __restrict__ dst, const float* __restrict__ ew,
                           float* __restrict__ deg, int E) {
    int e = blockIdx.x * 256 + threadIdx.x;
    if (e < E)
        __hip_atomic_fetch_add(&deg[dst[e]], ew[e],
                               __ATOMIC_RELAXED, __HIP_MEMORY_SCOPE_AGENT);
}

__global__ void rsqrt_kernel(float* __restrict__ d, int n) {
    int i = blockIdx.x * 256 + threadIdx.x;
    if (i < n) d[i] = rsqrtf(d[i]);
}

// One block per edge: agg[dst,:] += xw[src,:] * (dinv[src]*ew*dinv[dst])
__global__ __launch_bounds__(256)
void edge_agg_kernel(const int* __restrict__ src, const int* __restrict__ dst,
                     const float* __restrict__ ew, const float* __restrict__ dinv,
                     const float* __restrict__ xw, float* __restrict__ agg) {
    int e = blockIdx.x;
    int s = src[e], d = dst[e];
    float nrm = dinv[s] * ew[e] * dinv[d];
    const float* xr = xw + (size_t)s * 1024;
    float* ar = agg + (size_t)d * 1024;
    for (int f = threadIdx.x; f < 1024; f += 256)
        __hip_atomic_fetch_add(&ar[f], xr[f] * nrm,
                               __ATOMIC_RELAXED, __HIP_MEMORY_SCOPE_AGENT);
}

// node_out = leaky(agg + xw*dinv^2 + bias)  (in place on agg)
__global__ void gcn_finalize_kernel(float* __restrict__ agg, const float* __restrict__ xw,
                                    const float* __restrict__ dinv,
                                    const float* __restrict__ bias) {
    int f = blockIdx.x * 256 + threadIdx.x;   // gridDim.x = 4 -> f in [0,1024)
    int n = blockIdx.y;
    float di = dinv[n];
    size_t idx = (size_t)n * 1024 + f;
    float x = agg[idx] + xw[idx] * di * di + bias[f];
    agg[idx] = (x >= 0.f) ? x : 0.01f * x;
}

// Assemble feature [B, 4396] = [d_vecs | p_emb | nd[d_idx] | ns[p_idx] | nw[p_idx]]
__global__ void feature_kernel(const float* __restrict__ d_vecs, const float* __restrict__ p_emb,
                               const float* __restrict__ nd, const float* __restrict__ ns,
                               const float* __restrict__ nw, const int* __restrict__ d_idx,
                               const int* __restrict__ p_idx, float* __restrict__ feat) {
    int c = blockIdx.x * 256 + threadIdx.x;
    int b = blockIdx.y;
    if (c >= FDIM) return;
    float v;
    if (c < 300)       v = d_vecs[(size_t)b * 300 + c];
    else if (c < 1324) v = p_emb[(size_t)b * 1024 + (c - 300)];
    else if (c < 2348) v = nd[(size_t)d_idx[b] * 1024 + (c - 1324)];
    else if (c < 3372) v = ns[(size_t)p_idx[b] * 1024 + (c - 2348)];
    else               v = nw[(size_t)p_idx[b] * 1024 + (c - 3372)];
    feat[(size_t)b * FDIM + c] = v;
}

// BatchNorm stats (mean, biased var) over rows; 64 columns per block (coalesced)
__global__ __launch_bounds__(256)
void bn_stats_kernel(const float* __restrict__ h, float* __restrict__ mean,
                     float* __restrict__ var, int Bn, int C) {
    int cl = threadIdx.x & 63;
    int rl = threadIdx.x >> 6;          // 0..3
    int c  = blockIdx.x * 64 + cl;
    float s = 0.f, q = 0.f;
    if (c < C) {
        for (int r = rl; r < Bn; r += 4) {
            float v = h[(size_t)r * C + c];
            s += v; q += v * v;
        }
    }
    __shared__ float rs[256], rq[256];
    rs[threadIdx.x] = s; rq[threadIdx.x] = q;
    __syncthreads();
    if (rl == 0 && c < C) {
        float S = rs[cl] + rs[cl + 64] + rs[cl + 128] + rs[cl + 192];
        float Q = rq[cl] + rq[cl + 64] + rq[cl + 128] + rq[cl + 192];
        float m = S / (float)Bn;
        mean[c] = m;
        var[c]  = Q / (float)Bn - m * m;
    }
}

// out = act(g*(h-mean)*rsqrt(var+eps)+beta); act: 0 = relu, 1 = leaky(0.01)
__global__ void bn_apply_kernel(const float* __restrict__ h, const float* __restrict__ g,
                                const float* __restrict__ be, const float* __restrict__ mean,
                                const float* __restrict__ var, float* __restrict__ out,
                                int C, int leaky) {
    int c = blockIdx.x * 256 + threadIdx.x;
    int r = blockIdx.y;
    if (c >= C) return;
    size_t idx = (size_t)r * C + c;
    float x = g[c] * (h[idx] - mean[c]) * rsqrtf(var[c] + GEPS) + be[c];
    out[idx] = leaky ? ((x >= 0.f) ? x : 0.01f * x) : fmaxf(x, 0.f);
}

// y[row] = dot(h[row,:256], Wo) + bo
__global__ __launch_bounds__(256)
void out2_kernel(const float* __restrict__ h, const float* __restrict__ Wo,
                 const float* __restrict__ bo, float* __restrict__ y) {
    __shared__ float red[256];
    int row = blockIdx.x;
    float s = h[(size_t)row * 256 + threadIdx.x] * Wo[threadIdx.x];
    red[threadIdx.x] = s;
    __syncthreads();
    for (int off = 128; off > 0; off >>= 1) {
        if (threadIdx.x < off) red[threadIdx.x] += red[threadIdx.x + off];
        __syncthreads();
    }
    if (threadIdx.x == 0) y[row] = red[0] + bo[0];
}

// ---------------------------------------------------------------------------
// Host orchestration
// ---------------------------------------------------------------------------
static inline dim3 gemm_grid(int M, int N) {
    return dim3((unsigned)((N + 127) / 128), (unsigned)((M + 63) / 64), 1);
}

extern "C" void kernel_launch(void* const* d_in, const int* in_sizes, int n_in,
                              void* d_out, int out_size, void* d_ws, size_t ws_size,
                              hipStream_t stream) {
    // ---- inputs (setup_inputs dict order, params leaves flattened in order)
    const int*   d_index = (const int*)  d_in[0];
    const int*   p_index = (const int*)  d_in[1];
    const float* d_vecs  = (const float*)d_in[2];
    const float* p_emb   = (const float*)d_in[3];
    const float* d_ecfps = (const float*)d_in[4];
    const int*   d_ei    = (const int*)  d_in[5];
    const float* d_ew    = (const float*)d_in[6];
    const float* p_gos   = (const float*)d_in[7];
    const int*   p_ei    = (const int*)  d_in[8];
    const float* p_ew    = (const float*)d_in[9];
    const int*   p_ei_sw = (const int*)  d_in[10];
    const float* p_ew_sw = (const float*)d_in[11];
    const float* Wgd  = (const float*)d_in[12]; const float* bgd  = (const float*)d_in[13];
    const float* Wgs  = (const float*)d_in[14]; const float* bgs  = (const float*)d_in[15];
    const float* Wgw  = (const float*)d_in[16]; const float* bgw  = (const float*)d_in[17];
    const float* We1  = (const float*)d_in[18]; const float* be1  = (const float*)d_in[19];
    const float* ge1  = (const float*)d_in[20]; const float* bb1  = (const float*)d_in[21];
    const float* We2  = (const float*)d_in[22]; const float* be2  = (const float*)d_in[23];
    const float* ge2  = (const float*)d_in[24]; const float* bb2  = (const float*)d_in[25];
    const float* Wd1  = (const float*)d_in[26]; const float* bd1  = (const float*)d_in[27];
    const float* gd1  = (const float*)d_in[28]; const float* bbd1 = (const float*)d_in[29];
    const float* Wd2  = (const float*)d_in[30]; const float* bd2  = (const float*)d_in[31];
    const float* gd2  = (const float*)d_in[32]; const float* bbd2 = (const float*)d_in[33];
    const float* Wo1  = (const float*)d_in[34]; const float* bo1  = (const float*)d_in[35];
    const float* go1  = (const float*)d_in[36]; const float* bbo1 = (const float*)d_in[37];
    const float* Wo2  = (const float*)d_in[38]; const float* bo2  = (const float*)d_in[39];

    // ---- outputs: (y [B], decoded [B*4396], feature [B*4396]) flat
    float* y_out    = (float*)d_out;
    float* dec_out  = y_out + BDIM;
    float* feat_out = dec_out + (size_t)BDIM * FDIM;

    // ---- workspace carve-out
    float* ws  = (float*)d_ws;
    size_t off = 0;
    auto alloc = [&](size_t n) { float* p = ws + off; off += (n + 255) & ~(size_t)255; return p; };
    float* xw   = alloc((size_t)NNODE * 1024);  // per-conv X@W, reused 3x
    float* nod  = alloc((size_t)NNODE * 1024);  // drug node features (agg -> leaky out)
    float* nos  = alloc((size_t)NNODE * 1024);  // protein sis
    float* nsw  = alloc((size_t)NNODE * 1024);  // protein sw
    float* dinv = alloc(NNODE);                 // deg -> rsqrt, reused per conv
    float* h1   = alloc((size_t)BDIM * 2048);   // enc1 out, reused as dec1 out
    float* enc  = alloc((size_t)BDIM * 1024);   // enc2 out
    float* h4   = alloc((size_t)BDIM * FDIM);   // dec2 pre-BN
    float* ho1  = alloc((size_t)BDIM * 256);    // out1
    float* mean = alloc(FDIM);
    float* var  = alloc(FDIM);
    (void)ws_size; (void)n_in; (void)in_sizes; (void)out_size;

    const int nodeElems = NNODE * 1024;
    dim3 b256(256);

    // ---- one GCN conv: node_out = leaky(GCNConv(x, ei, ew; W, b)) ----------
    auto gcn = [&](const float* x, const int* ei, const float* ew,
                   const float* W, const float* bias, float* node_out) {
        hipLaunchKernelGGL(wmma_gemm_f32, gemm_grid(NNODE, 1024), b256, 0, stream,
                           x, W, (const float*)nullptr, xw, NNODE, 1024, 1024);
        hipLaunchKernelGGL(fill_kernel, dim3((nodeElems + 255) / 256), b256, 0, stream,
                           node_out, 0.f, nodeElems);
        hipLaunchKernelGGL(fill_kernel, dim3((NNODE + 255) / 256), b256, 0, stream,
                           dinv, 1.f, NNODE);                       // self-loop weight 1
        hipLaunchKernelGGL(deg_kernel, dim3((NEDGE + 255) / 256), b256, 0, stream,
                           ei + NEDGE, ew, dinv, NEDGE);
        hipLaunchKernelGGL(rsqrt_kernel, dim3((NNODE + 255) / 256), b256, 0, stream,
                           dinv, NNODE);
        hipLaunchKernelGGL(edge_agg_kernel, dim3(NEDGE), b256, 0, stream,
                           ei, ei + NEDGE, ew, dinv, xw, node_out);
        hipLaunchKernelGGL(gcn_finalize_kernel, dim3(4, NNODE), b256, 0, stream,
                           node_out, xw, dinv, bias);
    };

    gcn(d_ecfps, d_ei,    d_ew,    Wgd, bgd, nod);
    gcn(p_gos,   p_ei,    p_ew,    Wgs, bgs, nos);
    gcn(p_gos,   p_ei_sw, p_ew_sw, Wgw, bgw, nsw);

    // ---- feature concat straight into d_out slot ---------------------------
    hipLaunchKernelGGL(feature_kernel, dim3((FDIM + 255) / 256, BDIM), b256, 0, stream,
                       d_vecs, p_emb, nod, nos, nsw, d_index, p_index, feat_out);

    // ---- Linear -> BN -> act block -----------------------------------------
    auto block = [&](const float* A, const float* W, const float* bias,
                     const float* g, const float* beta, float* h, float* out,
                     int K, int N, int leaky) {
        hipLaunchKernelGGL(wmma_gemm_f32, gemm_grid(BDIM, N), b256, 0, stream,
                           A, W, bias, h, BDIM, N, K);
        hipLaunchKernelGGL(bn_stats_kernel, dim3((N + 63) / 64), b256, 0, stream,
                           h, mean, var, BDIM, N);
        hipLaunchKernelGGL(bn_apply_kernel, dim3((N + 255) / 256, BDIM), b256, 0, stream,
                           h, g, beta, mean, var, out, N, leaky);
    };

    block(feat_out, We1, be1, ge1, bb1,  h1,  h1,      FDIM, 2048, 0); // enc1 ReLU
    block(h1,       We2, be2, ge2, bb2,  enc, enc,     2048, 1024, 0); // enc2 ReLU
    block(enc,      Wd1, bd1, gd1, bbd1, h1,  h1,      1024, 2048, 0); // dec1 ReLU (reuse h1)
    block(h1,       Wd2, bd2, gd2, bbd2, h4,  dec_out, 2048, FDIM, 0); // dec2 ReLU -> d_out
    block(enc,      Wo1, bo1, go1, bbo1, ho1, ho1,     1024, 256,  1); // out1 Leaky

    hipLaunchKernelGGL(out2_kernel, dim3(BDIM), b256, 0, stream, ho1, Wo2, bo2, y_out);
}